// Embedder_29197187678536
// MI455X (gfx1250) — compile-verified
//
#include <hip/hip_runtime.h>

// Problem constants (match reference)
#define B_      16
#define T_      4096
#define C_      1024
#define T2_     (T_ + 2)          // 4098
#define PADIDX_ 1
#define THREADS 256               // 8 wave32s; 256 * float4 = 1024 = C

#if defined(__AMDGCN__) && __has_builtin(__builtin_amdgcn_global_load_async_to_lds_b128)
#define USE_ASYNC_LDS 1
#else
#define USE_ASYNC_LDS 0
#endif

#if USE_ASYNC_LDS
typedef int v4i_ __attribute__((vector_size(16)));                 // int vector[4]
typedef __attribute__((address_space(1))) v4i_* g_v4i_ptr;         // global (AS1)
typedef __attribute__((address_space(3))) v4i_* l_v4i_ptr;         // LDS (AS3)
#endif

__global__ __launch_bounds__(THREADS) void
embed_posln_kernel(const float* __restrict__ x,
                   const int*   __restrict__ lengths,
                   const float* __restrict__ bos,
                   const float* __restrict__ eos,
                   const float* __restrict__ pos_table,
                   const float* __restrict__ gamma,
                   const float* __restrict__ beta,
                   float*       __restrict__ out)
{
    __shared__ float s_x[C_];
    __shared__ float s_p[C_];
    __shared__ float s_red[16];     // 8 partial sums + 8 partial sumsq
    __shared__ float s_stats[2];    // mean, rstd

    const int row = blockIdx.x;          // 0 .. B*T2-1
    const int b   = row / T2_;
    const int t   = row - b * T2_;
    const int tid = threadIdx.x;
    const int off = tid * 4;             // float index within the row

    const int len = lengths[b];
    const int nl  = len + 2;

    // ---- pick x-row source (block-uniform branch; EXEC stays all-ones) ----
    const float* xsrc = nullptr;
    bool zero_row = false;
    if (t == 0)              xsrc = bos;
    else if (t == len + 1)   xsrc = eos;                                 // len+1 in [2, T]
    else if (t <= T_)        xsrc = x + ((size_t)b * T_ + (t - 1)) * (size_t)C_;
    else                     zero_row = true;                            // t == T+1 tail slot

    // position index: nonpad -> t+2, pad -> PADIDX (zeroed row of the table)
    const int    pidx = (t < nl) ? (t + 2) : PADIDX_;
    const float* psrc = pos_table + (size_t)pidx * (size_t)C_;

    float4 xv, pv;
#if USE_ASYNC_LDS
    // CDNA5 async data mover: each lane copies its own 16B into LDS (ASYNCcnt).
    if (!zero_row) {
        __builtin_amdgcn_global_load_async_to_lds_b128(
            (g_v4i_ptr)(xsrc + off), (l_v4i_ptr)(&s_x[off]), 0, 0);
    }
    __builtin_amdgcn_global_load_async_to_lds_b128(
        (g_v4i_ptr)(psrc + off), (l_v4i_ptr)(&s_p[off]), 0, 0);
#if __has_builtin(__builtin_amdgcn_s_wait_asynccnt)
    __builtin_amdgcn_s_wait_asynccnt(0);
#else
    asm volatile("s_wait_asynccnt 0" ::: "memory");
#endif
    asm volatile("" ::: "memory");  // keep LDS reads below the wait
    pv = *reinterpret_cast<const float4*>(&s_p[off]);
    if (zero_row) xv = make_float4(0.f, 0.f, 0.f, 0.f);
    else          xv = *reinterpret_cast<const float4*>(&s_x[off]);
#else
    pv = *reinterpret_cast<const float4*>(psrc + off);
    if (zero_row) xv = make_float4(0.f, 0.f, 0.f, 0.f);
    else          xv = *reinterpret_cast<const float4*>(xsrc + off);
#endif

    // ---- y = x + pos; accumulate sum / sumsq ----
    float4 y;
    y.x = xv.x + pv.x;  y.y = xv.y + pv.y;
    y.z = xv.z + pv.z;  y.w = xv.w + pv.w;

    float s  = y.x + y.y + y.z + y.w;
    float s2 = y.x * y.x + y.y * y.y + y.z * y.z + y.w * y.w;

    // ---- wave32 butterfly reduction ----
    #pragma unroll
    for (int m = 16; m >= 1; m >>= 1) {
        s  += __shfl_xor(s,  m, 32);
        s2 += __shfl_xor(s2, m, 32);
    }
    const int wave = tid >> 5;
    if ((tid & 31) == 0) { s_red[wave] = s; s_red[8 + wave] = s2; }
    __syncthreads();

    if (tid == 0) {
        float ts = 0.f, ts2 = 0.f;
        #pragma unroll
        for (int w = 0; w < 8; ++w) { ts += s_red[w]; ts2 += s_red[8 + w]; }
        const float mean = ts * (1.0f / C_);
        float var = ts2 * (1.0f / C_) - mean * mean;
        var = fmaxf(var, 0.0f);
        s_stats[0] = mean;
        s_stats[1] = rsqrtf(var + 1e-5f);
    }
    __syncthreads();

    const float mean = s_stats[0];
    const float rstd = s_stats[1];

    // ---- normalize + affine, 128-bit store ----
    const float4 g  = *reinterpret_cast<const float4*>(gamma + off);
    const float4 bt = *reinterpret_cast<const float4*>(beta  + off);
    float4 o;
    o.x = (y.x - mean) * rstd * g.x + bt.x;
    o.y = (y.y - mean) * rstd * g.y + bt.y;
    o.z = (y.z - mean) * rstd * g.z + bt.z;
    o.w = (y.w - mean) * rstd * g.w + bt.w;
    *reinterpret_cast<float4*>(out + (size_t)row * (size_t)C_ + off) = o;

    // ---- auxiliary outputs: pad_mask (float 0/1) and new_lengths (float) ----
    if (tid == 0) {
        const size_t y_elems = (size_t)B_ * T2_ * C_;
        out[y_elems + (size_t)row] = (t >= nl) ? 1.0f : 0.0f;
        if (t == 0)
            out[y_elems + (size_t)B_ * T2_ + (size_t)b] = (float)nl;
    }
}

extern "C" void kernel_launch(void* const* d_in, const int* in_sizes, int n_in,
                              void* d_out, int out_size, void* d_ws, size_t ws_size,
                              hipStream_t stream)
{
    (void)in_sizes; (void)n_in; (void)out_size; (void)d_ws; (void)ws_size;
    // setup_inputs order: x, padding_mask(unused), lengths, bos_emb, eos_emb,
    //                     pos_table, ln_weight, ln_bias
    const float* x        = (const float*)d_in[0];
    const int*   lengths  = (const int*)  d_in[2];
    const float* bos      = (const float*)d_in[3];
    const float* eos      = (const float*)d_in[4];
    const float* postab   = (const float*)d_in[5];
    const float* gamma    = (const float*)d_in[6];
    const float* beta     = (const float*)d_in[7];
    float*       out      = (float*)d_out;

    dim3 grid(B_ * T2_);   // 65568 rows, one block each
    dim3 block(THREADS);
    embed_posln_kernel<<<grid, block, 0, stream>>>(x, lengths, bos, eos,
                                                   postab, gamma, beta, out);
}